// ChamferLoss_8495445311604
// MI455X (gfx1250) — compile-verified
//
#include <hip/hip_runtime.h>

typedef __attribute__((ext_vector_type(2))) float v2f;
typedef __attribute__((ext_vector_type(8))) float v8f;

#define NPTS 8192
#define TILE 16
#define ROWS_PER_WAVE 32   // two 16-row strips per wave
#define WAVES_PER_BLOCK 8

__global__ void chamfer_zero_out(float* out) { out[0] = 0.0f; }

// Computes  sum_i min_j ||x_i - y_j||^2  over all batches, accumulated
// atomically into out[0].  x, y are [B, 3, NPTS] coordinate planes.
//
// One wave32 owns 32 rows of x (two 16-row strips, two loop-invariant A
// operands) and sweeps all 512 column tiles of y.  Each 16x16 squared-
// distance tile comes from a single v_wmma_f32_16x16x4_f32:
//   A[m][0..2] = -2*x_m,  A[m][3] = 1.0
//   B[0..2][n] =  y_n,    B[3][n] = |y_n|^2
//   C          =  rx[m]   (loop-invariant)
// so D[m][n] = rx + ry - 2*x.y = ||x_m - y_n||^2 with no per-tile C setup
// and no cross-lane shuffles in the hot loop.
__global__ __launch_bounds__(256) void chamfer_rowmin_sum(
    const float* __restrict__ x, const float* __restrict__ y,
    float* __restrict__ out) {
  const int lane = threadIdx.x & 31;
  const int wave = threadIdx.x >> 5;
  const int task = blockIdx.x * WAVES_PER_BLOCK + wave;
  const int tasksPerBatch = NPTS / ROWS_PER_WAVE;  // 256
  const int b  = task / tasksPerBatch;
  const int i0 = (task % tasksPerBatch) * ROWS_PER_WAVE;

  const float* xb = x + b * 3 * NPTS;
  const float* yb = y + b * 3 * NPTS;

  const int  m  = lane & 15;
  const bool hi = lane >= 16;

  // ---- Two loop-invariant A tiles (rows i0+m and i0+16+m).
  // 32-bit A 16x4 layout: lanes 0-15 hold (K=0,K=1), lanes 16-31 hold (K=2,K=3).
  const float x00 = xb[0 * NPTS + i0 + m];
  const float x01 = xb[1 * NPTS + i0 + m];
  const float x02 = xb[2 * NPTS + i0 + m];
  const float x10 = xb[0 * NPTS + i0 + 16 + m];
  const float x11 = xb[1 * NPTS + i0 + 16 + m];
  const float x12 = xb[2 * NPTS + i0 + 16 + m];

  v2f a0, a1;
  a0.x = -2.0f * (hi ? x02 : x00);
  a0.y = hi ? 1.0f : (-2.0f * x01);   // K=3 slot multiplies B's ry row
  a1.x = -2.0f * (hi ? x12 : x10);
  a1.y = hi ? 1.0f : (-2.0f * x11);

  // ---- Row squared norms as loop-invariant C operands (C/D row m = v + 8*hi).
  const float rn0 = x00 * x00 + x01 * x01 + x02 * x02;
  const float rn1 = x10 * x10 + x11 * x11 + x12 * x12;
  v8f rx0, rx1;
#pragma unroll
  for (int v = 0; v < 8; ++v) {
    rx0[v] = __shfl(rn0, v + (hi ? 8 : 0), 32);
    rx1[v] = __shfl(rn1, v + (hi ? 8 : 0), 32);
  }

  v8f rm0, rm1;
#pragma unroll
  for (int v = 0; v < 8; ++v) {
    rm0[v] = __builtin_inff();
    rm1[v] = __builtin_inff();
  }

#pragma unroll 2
  for (int j0 = 0; j0 < NPTS; j0 += TILE) {
    // B tile: every lane loads all 3 coords of point j0 + (lane&15);
    // 16-lane-coalesced 64B reads per plane, all cache-resident.
    const float y0 = yb[0 * NPTS + j0 + m];
    const float y1 = yb[1 * NPTS + j0 + m];
    const float y2 = yb[2 * NPTS + j0 + m];
    const float ry = y0 * y0 + y1 * y1 + y2 * y2;

    v2f bm;
    bm.x = hi ? y2 : y0;  // lanes 0-15: K=0 ; lanes 16-31: K=2
    bm.y = hi ? ry : y1;  // lanes 0-15: K=1 ; lanes 16-31: K=3 (= |y_n|^2)

    v8f d0 = __builtin_amdgcn_wmma_f32_16x16x4_f32(
        false, a0, false, bm, (short)0, rx0, false, false);
    v8f d1 = __builtin_amdgcn_wmma_f32_16x16x4_f32(
        false, a1, false, bm, (short)0, rx1, false, false);

#pragma unroll
    for (int v = 0; v < 8; ++v) {
      rm0[v] = fminf(rm0[v], d0[v]);
      rm1[v] = fminf(rm1[v], d1[v]);
    }
  }

  // ---- Row-min reduction across the 16 lanes of each half-wave (tail only).
#pragma unroll
  for (int v = 0; v < 8; ++v) {
    float r0 = rm0[v];
    float r1 = rm1[v];
    r0 = fminf(r0, __shfl_xor(r0, 1, 32));
    r1 = fminf(r1, __shfl_xor(r1, 1, 32));
    r0 = fminf(r0, __shfl_xor(r0, 2, 32));
    r1 = fminf(r1, __shfl_xor(r1, 2, 32));
    r0 = fminf(r0, __shfl_xor(r0, 4, 32));
    r1 = fminf(r1, __shfl_xor(r1, 4, 32));
    r0 = fminf(r0, __shfl_xor(r0, 8, 32));
    r1 = fminf(r1, __shfl_xor(r1, 8, 32));
    rm0[v] = r0;  // lanes 0-15: min of row v; lanes 16-31: min of row v+8
    rm1[v] = r1;
  }
  float partial = 0.0f;
#pragma unroll
  for (int v = 0; v < 8; ++v) partial += rm0[v] + rm1[v];
  partial += __shfl_xor(partial, 16, 32);  // rows 0-7 + rows 8-15 halves

  if (lane == 0) atomicAdd(out, partial);
}

extern "C" void kernel_launch(void* const* d_in, const int* in_sizes, int n_in,
                              void* d_out, int out_size, void* d_ws,
                              size_t ws_size, hipStream_t stream) {
  const float* preds = (const float*)d_in[0];  // [8, 3, 8192]
  const float* gts   = (const float*)d_in[1];  // [8, 3, 8192]
  float* out = (float*)d_out;                  // scalar loss

  const int B = 8;
  const int blocks = B * (NPTS / ROWS_PER_WAVE) / WAVES_PER_BLOCK;  // 256

  chamfer_zero_out<<<1, 1, 0, stream>>>(out);
  // loss_2: for each pred, nearest gt (row-mins of P(preds, gts))
  chamfer_rowmin_sum<<<blocks, 256, 0, stream>>>(preds, gts, out);
  // loss_1: for each gt, nearest pred (row-mins of P(gts, preds))
  chamfer_rowmin_sum<<<blocks, 256, 0, stream>>>(gts, preds, out);
}